// UMAPNEGLoss_42150809043634
// MI455X (gfx1250) — compile-verified
//
#include <hip/hip_runtime.h>

typedef __attribute__((ext_vector_type(2))) float v2f;
typedef __attribute__((ext_vector_type(8))) float v8f;

#define N_NEG 20
#define EPSF 1e-12f

// One wave handles 16 batch rows. For each of the 21 neighbors it builds the
// 16x64 difference matrix W in registers (lane = row (L&15), element pairs of
// parity (L>>4), matching the f32 16x16x4 WMMA A/B striping) and computes
// C = W * W^T with 16 chained v_wmma_f32_16x16x4_f32. diag(C) = squared dists.
__global__ __launch_bounds__(256) void umap_neg_loss_kernel(
    const float* __restrict__ logits,
    const int*   __restrict__ xs,
    const int*   __restrict__ ys,
    const int*   __restrict__ y_neg,
    const float* __restrict__ w_arr,
    float*       __restrict__ partial,
    int Btotal)
{
    __shared__ float red[256];
    const int lane = threadIdx.x & 31;
    const int wave = threadIdx.x >> 5;
    const int tile = blockIdx.x * 8 + wave;

    float acc = 0.0f;
    if (tile * 16 < Btotal) {                 // wave-uniform branch: EXEC all-1s inside
        const int b0 = tile * 16;
        const int r  = lane & 15;             // matrix row this lane feeds
        const int h  = lane >> 4;             // which pair-parity of the 64 elems
        int brow = b0 + r;
        if (brow >= Btotal) brow = Btotal - 1; // clamp (tail safety)
        const bool rowOK = (b0 + r) < Btotal;

        const float w   = w_arr[brow];
        const float wm1 = w - 1.0f;

        // gather this row's x-vector in WMMA striping: 16 float2 chunks
        const size_t xbase = (size_t)xs[brow] * 64 + 2 * h;
        v2f xv[16];
#pragma unroll
        for (int k = 0; k < 16; ++k)
            xv[k] = *(const v2f*)(logits + xbase + 4 * k);

        // diagonal element C[j][j]: lane j holds it in VGPR j (j<8),
        // lane j+16 holds it in VGPR j-8 (j>=8)
        const int  vi    = (lane < 16) ? lane : (lane - 24);
        const bool valid = rowOK && ((lane < 8) || (lane >= 24));

        for (int t = 0; t < 21; ++t) {
            const int nidx = (t == 0) ? ys[brow] : y_neg[brow * N_NEG + (t - 1)];
            const size_t vbase = (size_t)nidx * 64 + 2 * h;

            v8f c = {0.f, 0.f, 0.f, 0.f, 0.f, 0.f, 0.f, 0.f};
#pragma unroll
            for (int k = 0; k < 16; ++k) {
                v2f vv = *(const v2f*)(logits + vbase + 4 * k);
                v2f d  = xv[k] - vv;
                // C += D_k * D_k^T  (A and B striping coincide for W*W^T)
                c = __builtin_amdgcn_wmma_f32_16x16x4_f32(
                        false, d, false, d, (short)0, c, false, false);
            }

            float dis = c[0];
            dis = (vi == 1) ? c[1] : dis;
            dis = (vi == 2) ? c[2] : dis;
            dis = (vi == 3) ? c[3] : dis;
            dis = (vi == 4) ? c[4] : dis;
            dis = (vi == 5) ? c[5] : dis;
            dis = (vi == 6) ? c[6] : dis;
            dis = (vi == 7) ? c[7] : dis;

            const float q = 1.0f / (1.0f + 0.25f * dis);
            float term;
            if (t == 0) {
                float p = fminf(fmaxf(q, EPSF), 1.0f - EPSF);
                term = w * logf(w / p) * (float)N_NEG;
            } else {
                float p = fminf(fmaxf(q, EPSF), 0.99f);
                term = wm1 * logf((p - 1.0f) / wm1);   // both factors negative -> finite
            }
            acc += valid ? term : 0.0f;
        }
    }

    red[threadIdx.x] = acc;
    __syncthreads();
    if (threadIdx.x == 0) {
        float s = 0.0f;
        for (int i = 0; i < 256; ++i) s += red[i];   // fixed order: deterministic
        partial[blockIdx.x] = s;
    }
}

__global__ void umap_final_reduce_kernel(const float* __restrict__ partial,
                                         float* __restrict__ out, int n)
{
    if (threadIdx.x == 0 && blockIdx.x == 0) {
        float s = 0.0f;
        for (int i = 0; i < n; ++i) s += partial[i]; // fixed order: deterministic
        out[0] = s;
    }
}

extern "C" void kernel_launch(void* const* d_in, const int* in_sizes, int n_in,
                              void* d_out, int out_size, void* d_ws, size_t ws_size,
                              hipStream_t stream) {
    const float* logits = (const float*)d_in[0];
    const int*   xs     = (const int*)  d_in[1];
    const int*   ys     = (const int*)  d_in[2];
    const int*   y_neg  = (const int*)  d_in[3];
    const float* w      = (const float*)d_in[4];

    const int B      = in_sizes[1];          // xs length = batch size
    const int tiles  = (B + 15) / 16;
    const int blocks = (tiles + 7) / 8;      // 8 waves (tiles) per 256-thread block

    float* partial = (float*)d_ws;           // blocks * 4 bytes of scratch

    umap_neg_loss_kernel<<<blocks, 256, 0, stream>>>(
        logits, xs, ys, y_neg, w, partial, B);
    umap_final_reduce_kernel<<<1, 1, 0, stream>>>(partial, (float*)d_out, blocks);
}